// Mult_1580547969011
// MI455X (gfx1250) — compile-verified
//
#include <hip/hip_runtime.h>

typedef __attribute__((ext_vector_type(2))) float v2f;
typedef __attribute__((ext_vector_type(8))) float v8f;

namespace {
constexpr int kNC  = 64;
constexpr int kHW  = 128 * 128;
constexpr int kB   = 32;
constexpr int kPos = kB * kHW;             // 524288 GEMM rows (spatial positions)
constexpr int kTiles = kPos / 16;          // 32768 position tiles of 16
constexpr int kThreads = 256;              // 8 wave32 per block
constexpr int kWavesPerBlock = kThreads / 32;
}

// out[b,j,p] = exp2( sum_i log2(relu(x[b,i,p])+0.1) * E[i,j] ) + bias[j]
// GEMM roles (transposed tile for coalesced channel-major stores):
//   A (16x4,  MxK) = E^T slice   : M = output channel, K = input channel (from LDS)
//   B (4x16,  KxN) = log2-x tile : K = input channel, N = 16 consecutive positions
//   D (16x16, MxN) -> out, VGPR r / lane l maps to channel j0+r+8*(l>>4), pos p+(l&15)
__global__ __launch_bounds__(kThreads) void powmix_wmma_kernel(
    const float* __restrict__ x, const float* __restrict__ E,
    const float* __restrict__ bias, float* __restrict__ out) {
  __shared__ float ldsE[kNC * kNC];   // E[i*64 + j], 16 KB
  __shared__ float ldsBias[kNC];
  for (int i = threadIdx.x; i < kNC * kNC; i += kThreads) ldsE[i] = E[i];
  if (threadIdx.x < kNC) ldsBias[threadIdx.x] = bias[threadIdx.x];
  __syncthreads();

  const int lane   = threadIdx.x & 31;
  const int laneLo = lane & 15;   // A: row M      B/D: col N (position)
  const int laneHi = lane >> 4;   // selects K pair (A/B) and M-half (D)
  const int wave   = threadIdx.x >> 5;
  const int numWaves = gridDim.x * kWavesPerBlock;

  for (int t = blockIdx.x * kWavesPerBlock + wave; t < kTiles; t += numWaves) {
    const int g = t << 4;           // tile never crosses a batch (HW % 16 == 0)
    const int b = g / kHW;
    const int p = g - b * kHW;

    // ---- load 64(K) x 16(N) x-tile, relu+0.1, native log2 -> B fragments ----
    // lane reads channels 4*kg + 2*laneHi + {0,1} at position p+laneLo:
    // each global_load_b32 pulls two contiguous 64B runs.
    const float* xb = x + (size_t)b * kNC * kHW + (size_t)(2 * laneHi) * kHW
                        + (size_t)(p + laneLo);
    v2f Bt[16];
#pragma unroll
    for (int kg = 0; kg < 16; ++kg) {
      float b0 = xb[(size_t)(4 * kg)     * kHW];
      float b1 = xb[(size_t)(4 * kg + 1) * kHW];
      b0 = fmaxf(b0, 0.0f) + 0.1f;
      b1 = fmaxf(b1, 0.0f) + 0.1f;
      Bt[kg].x = __builtin_amdgcn_logf(b0);   // v_log_f32 (log2; ln2*log2e cancels)
      Bt[kg].y = __builtin_amdgcn_logf(b1);
    }

    const float* eBase  = ldsE + (2 * laneHi) * kNC + laneLo;
    float* outBase = out + (size_t)b * kNC * kHW + (size_t)(8 * laneHi) * kHW
                         + (size_t)(p + laneLo);

#pragma unroll
    for (int jt = 0; jt < 4; ++jt) {          // 4 output-channel tiles of 16
      const int j0 = jt * 16;
      const float* ep = eBase + j0;           // A(m,k) = E[k][j0+m] from LDS
      v8f acc = {};
#pragma unroll
      for (int kg = 0; kg < 16; ++kg) {       // K = 64 in steps of 4
        v2f At;
        At.x = ep[(4 * kg)     * kNC];
        At.y = ep[(4 * kg + 1) * kNC];
        acc = __builtin_amdgcn_wmma_f32_16x16x4_f32(
            /*neg_a=*/false, At, /*neg_b=*/false, Bt[kg],
            /*c_mod=*/(short)0, acc, /*reuse_a=*/false, /*reuse_b=*/false);
      }
#pragma unroll
      for (int r = 0; r < 8; ++r) {           // exp2, +bias, contiguous stores
        float v = __builtin_amdgcn_exp2f(acc[r]) + ldsBias[j0 + 8 * laneHi + r];
        outBase[(size_t)(j0 + r) * kHW] = v;
      }
    }
  }
}

extern "C" void kernel_launch(void* const* d_in, const int* in_sizes, int n_in,
                              void* d_out, int out_size, void* d_ws, size_t ws_size,
                              hipStream_t stream) {
  const float* x    = (const float*)d_in[0];
  const float* E    = (const float*)d_in[1];   // exp_param, trailing 1x1 dims flat
  const float* bias = (const float*)d_in[2];
  float* out = (float*)d_out;
  dim3 grid(kTiles / kWavesPerBlock);   // 4096 blocks -> one tile per wave
  dim3 block(kThreads);
  hipLaunchKernelGGL(powmix_wmma_kernel, grid, block, 0, stream, x, E, bias, out);
  (void)in_sizes; (void)n_in; (void)out_size; (void)d_ws; (void)ws_size;
}